// GNNModel_87385404605104
// MI455X (gfx1250) — compile-verified
//
#include <hip/hip_runtime.h>
#include <hip/hip_bf16.h>

typedef float v2f __attribute__((ext_vector_type(2)));
typedef float v8f __attribute__((ext_vector_type(8)));

#define NNODES 100000
#define NEDGES 1600000
#define NGRAPHS 64
#define HID 128
#define NEG_SLOPE 0.01f

// ---------------- utility: zero a region (float4 granular) ----------------
__global__ void zero_f4(float4* __restrict__ p, int n4) {
    int i = blockIdx.x * blockDim.x + threadIdx.x;
    if (i < n4) p[i] = make_float4(0.f, 0.f, 0.f, 0.f);
}

// ---------------- degree count: deg[dst] += 1 ----------------
__global__ void degree_kernel(const int* __restrict__ dst, float* __restrict__ deg, int ne) {
    int e = blockIdx.x * blockDim.x + threadIdx.x;
    if (e < ne) atomicAdd(&deg[dst[e]], 1.0f);
}

// ---------------- dinv = rsqrt(deg + 1)  (self loop) ----------------
__global__ void dinv_kernel(const float* __restrict__ deg, float* __restrict__ dinv, int n) {
    int i = blockIdx.x * blockDim.x + threadIdx.x;
    if (i < n) dinv[i] = rsqrtf(deg[i] + 1.0f);
}

// ---------------- fp32 WMMA GEMM: C[M,128] = A[M,K] @ B[K,128] ----------------
// grid.x = M/16 ; block = 256 threads = 8 waves ; wave w -> N-tile [16w, 16w+16)
// A tile (16 x K) staged in LDS with +4 float row padding (stride 260 mod 64 = 4
// -> conflict-free lane access).
template <int K>
__global__ __launch_bounds__(256) void gemm_wmma(const float* __restrict__ A,
                                                 const float* __restrict__ B,
                                                 float* __restrict__ C) {
    constexpr int LDA = K + 4;
    __shared__ float As[16 * LDA];

    const int tid    = threadIdx.x;
    const int tile_m = blockIdx.x;

    // cooperative load of 16 x K A-tile (float4 chunks)
    const float4* Ag = (const float4*)(A + (size_t)tile_m * 16 * K);
    for (int i = tid; i < 16 * (K / 4); i += 256) {
        int row = i / (K / 4);
        int c4  = i % (K / 4);
        *(float4*)&As[row * LDA + c4 * 4] = Ag[row * (K / 4) + c4];
    }
    __syncthreads();

    const int wave  = tid >> 5;        // 0..7 -> N tile
    const int lane  = tid & 31;
    const int n0    = wave * 16;
    const int row   = lane & 15;       // A row / B col within tile
    const int khalf = (lane >> 4) * 2; // 0 for lanes 0-15, 2 for lanes 16-31

    v8f acc = {};
    for (int k = 0; k < K; k += 4) {
        v2f a, b;
        a.x = As[row * LDA + k + khalf];
        a.y = As[row * LDA + k + khalf + 1];
        const float* Bp = B + (size_t)(k + khalf) * HID + n0 + row;
        b.x = Bp[0];
        b.y = Bp[HID];
        acc = __builtin_amdgcn_wmma_f32_16x16x4_f32(false, a, false, b,
                                                    (short)0, acc, false, false);
    }

    // C/D layout: VGPR r -> M = r (lanes 0-15) / r+8 (lanes 16-31), N = lane&15
    const int cn      = lane & 15;
    const int cm_base = (lane >> 4) * 8;
    float* Cp = C + ((size_t)tile_m * 16 + cm_base) * HID + n0 + cn;
#pragma unroll
    for (int r = 0; r < 8; ++r) Cp[(size_t)r * HID] = acc[r];
}

// ---------------- edge scatter: agg[dst] += h[src] * dinv[src]*dinv[dst] -------
// one wave32 per edge, float4 per lane (128 channels), fp32 L2 atomics
__global__ __launch_bounds__(256) void edge_kernel(const float* __restrict__ h,
                                                   const int* __restrict__ src,
                                                   const int* __restrict__ dst,
                                                   const float* __restrict__ dinv,
                                                   float* __restrict__ agg, int ne) {
    int e = blockIdx.x * (blockDim.x >> 5) + (threadIdx.x >> 5);
    if (e >= ne) return;
    int lane = threadIdx.x & 31;
    int s = src[e];
    int d = dst[e];
    float norm = dinv[s] * dinv[d];
    float4 v = ((const float4*)(h + (size_t)s * HID))[lane];
    float* ap = agg + (size_t)d * HID + lane * 4;
    atomicAdd(ap + 0, v.x * norm);
    atomicAdd(ap + 1, v.y * norm);
    atomicAdd(ap + 2, v.z * norm);
    atomicAdd(ap + 3, v.w * norm);
}

// ---------------- fused self-loop + bias + leaky ReLU (in place on agg) -------
__global__ void finish_kernel(const float* __restrict__ h, float* __restrict__ agg,
                              const float* __restrict__ dinv,
                              const float* __restrict__ bias, int n) {
    size_t i = (size_t)blockIdx.x * blockDim.x + threadIdx.x;
    if (i >= (size_t)n * HID) return;
    int node = (int)(i >> 7);
    int c    = (int)(i & (HID - 1));
    float di = dinv[node];
    float v  = agg[i] + h[i] * di * di + bias[c];
    agg[i]   = v > 0.f ? v : v * NEG_SLOPE;
}

// ---------------- pool: sums[batch] += h ; cnts[batch] += 1 ----------------
__global__ __launch_bounds__(256) void pool_kernel(const float* __restrict__ h,
                                                   const int* __restrict__ batch,
                                                   float* __restrict__ sums,
                                                   float* __restrict__ cnts, int n) {
    int node = blockIdx.x * (blockDim.x >> 5) + (threadIdx.x >> 5);
    if (node >= n) return;
    int lane = threadIdx.x & 31;
    int g = batch[node];
    float4 v = ((const float4*)(h + (size_t)node * HID))[lane];
    float* sp = sums + (size_t)g * HID + lane * 4;
    atomicAdd(sp + 0, v.x);
    atomicAdd(sp + 1, v.y);
    atomicAdd(sp + 2, v.z);
    atomicAdd(sp + 3, v.w);
    if (lane == 0) atomicAdd(&cnts[g], 1.0f);
}

// ---------------- final FC: logits[G,2] = (sums/cnt) @ Wfc + bfc ----------------
__global__ void fc_kernel(const float* __restrict__ sums, const float* __restrict__ cnts,
                          const float* __restrict__ Wfc, const float* __restrict__ bfc,
                          float* __restrict__ out) {
    int t = threadIdx.x;          // 0..127
    if (t >= NGRAPHS * 2) return;
    int g = t >> 1, o = t & 1;
    float inv = 1.0f / fmaxf(cnts[g], 1.0f);
    float acc = 0.f;
    for (int k = 0; k < HID; ++k) acc += sums[g * HID + k] * Wfc[k * 2 + o];
    out[t] = acc * inv + bfc[o];
}

extern "C" void kernel_launch(void* const* d_in, const int* in_sizes, int n_in,
                              void* d_out, int out_size, void* d_ws, size_t ws_size,
                              hipStream_t stream) {
    const float* x    = (const float*)d_in[0];            // [N,256]
    const int*   ei   = (const int*)d_in[1];              // [2,E]
    const int*   batch= (const int*)d_in[2];              // [N]
    const float* W1   = (const float*)d_in[3];            // [256,128]
    const float* b1   = (const float*)d_in[4];
    const float* W2   = (const float*)d_in[5];            // [128,128]
    const float* b2   = (const float*)d_in[6];
    const float* Wfc  = (const float*)d_in[7];            // [128,2]
    const float* bfc  = (const float*)d_in[8];
    float* out = (float*)d_out;                           // [64,2]

    const int* src = ei;
    const int* dst = ei + NEDGES;

    // workspace layout (bytes)
    char* ws = (char*)d_ws;
    const size_t featBytes = (size_t)NNODES * HID * sizeof(float);  // 51.2 MB
    float* buf0 = (float*)(ws);                    // h (gemm output)
    float* buf1 = (float*)(ws + featBytes);        // agg -> activated output (in place)
    float* deg  = (float*)(ws + 2 * featBytes);                    // [N]
    float* dinv = (float*)(ws + 2 * featBytes + NNODES * 4);       // [N]
    float* sums = (float*)(ws + 2 * featBytes + 2 * NNODES * 4);   // [64,128]
    float* cnts = sums + NGRAPHS * HID;                            // [64] (adjacent)

    const int featF4 = NNODES * HID / 4;          // 3.2M float4
    const int poolF4 = (NGRAPHS * HID + NGRAPHS) / 4;

    // --- init: zero deg, agg(buf1), pool sums+cnts ---
    zero_f4<<<(NNODES / 4 + 255) / 256, 256, 0, stream>>>((float4*)deg, NNODES / 4);
    zero_f4<<<(featF4 + 255) / 256, 256, 0, stream>>>((float4*)buf1, featF4);
    zero_f4<<<(poolF4 + 255) / 256, 256, 0, stream>>>((float4*)sums, poolF4);

    // --- degrees and symmetric norm ---
    degree_kernel<<<(NEDGES + 255) / 256, 256, 0, stream>>>(dst, deg, NEDGES);
    dinv_kernel<<<(NNODES + 255) / 256, 256, 0, stream>>>(deg, dinv, NNODES);

    // --- layer 1 ---
    gemm_wmma<256><<<NNODES / 16, 256, 0, stream>>>(x, W1, buf0);          // h1
    edge_kernel<<<NEDGES / 8, 256, 0, stream>>>(buf0, src, dst, dinv, buf1, NEDGES);
    finish_kernel<<<(NNODES * HID + 255) / 256, 256, 0, stream>>>(buf0, buf1, dinv, b1, NNODES);

    // --- layer 2 (buf1 now holds activated h; gemm into buf0, re-zero buf1 as agg) ---
    gemm_wmma<128><<<NNODES / 16, 256, 0, stream>>>(buf1, W2, buf0);       // h2
    zero_f4<<<(featF4 + 255) / 256, 256, 0, stream>>>((float4*)buf1, featF4);
    edge_kernel<<<NEDGES / 8, 256, 0, stream>>>(buf0, src, dst, dinv, buf1, NEDGES);
    finish_kernel<<<(NNODES * HID + 255) / 256, 256, 0, stream>>>(buf0, buf1, dinv, b2, NNODES);

    // --- pool + FC ---
    pool_kernel<<<(NNODES + 7) / 8, 256, 0, stream>>>(buf1, batch, sums, cnts, NNODES);
    fc_kernel<<<1, 128, 0, stream>>>(sums, cnts, Wfc, bfc, out);
}